// SAGELayer_14224931684660
// MI455X (gfx1250) — compile-verified
//
#include <hip/hip_runtime.h>

typedef float v2f __attribute__((ext_vector_type(2)));
typedef float v8f __attribute__((ext_vector_type(8)));

#define D_FEAT 128
#define D_OUT 128
#define K_TOTAL 256          // 2 * D_FEAT
#define KC 64                // K-chunk staged in LDS
#define TSTRIDE 68           // dwords per transposed-LDS column (64 + 4 pad):
                             // compute b64 reads hit all 64 banks exactly once

// ---------------------------------------------------------------------------
// Kernel 1: zero msg_sum + deg workspace
// ---------------------------------------------------------------------------
__global__ void sage_zero_kernel(float* __restrict__ p, int n) {
  int i = blockIdx.x * blockDim.x + threadIdx.x;
  if (i < n) p[i] = 0.0f;
}

// ---------------------------------------------------------------------------
// Kernel 2: edge scatter. One wave per edge: 32 lanes x float4 = 128 floats.
// Gather features[src] (coalesced 512B), atomic-add into msg[dst] (L2-resident),
// lane 0 bumps the degree counter.
// ---------------------------------------------------------------------------
__global__ void sage_scatter_kernel(const float* __restrict__ feat,
                                    const int* __restrict__ src,
                                    const int* __restrict__ dst,
                                    float* __restrict__ msg,
                                    float* __restrict__ deg,
                                    int nEdges) {
  int gid  = blockIdx.x * blockDim.x + threadIdx.x;
  int edge = gid >> 5;
  int lane = threadIdx.x & 31;
  if (edge >= nEdges) return;

  int s = __builtin_amdgcn_readfirstlane(src[edge]);
  int d = __builtin_amdgcn_readfirstlane(dst[edge]);

  const float4* fp = (const float4*)(feat + (size_t)s * D_FEAT);
  float4 v = fp[lane];

  float* mp = msg + (size_t)d * D_FEAT + lane * 4;
  atomicAdd(mp + 0, v.x);
  atomicAdd(mp + 1, v.y);
  atomicAdd(mp + 2, v.z);
  atomicAdd(mp + 3, v.w);
  if (lane == 0) atomicAdd(deg + d, 1.0f);
}

// ---------------------------------------------------------------------------
// Kernel 3: out = concat(feat, msg/max(deg,1)) @ W + b  via V_WMMA_F32_16X16X4_F32
// Block = 256 threads (8 waves). Wave w owns rows [blockIdx*128 + w*16, +16)
// across all 128 output columns (8 WMMA N-tiles, v8f each).
// W is staged TRANSPOSED through LDS (Wt[n][k], stride 68) so each B fragment
// (W[kb][n], W[kb+1][n]) is one bank-conflict-free ds_load_b64.
// Inner loop batches: per 2 k-steps, ALL 16 B fragments + 2 A fragments are
// loaded into registers first, then a 16-WMMA burst runs — so LDS/global
// latency is amortized across the burst instead of stalling every WMMA pair.
// ---------------------------------------------------------------------------
__global__ __launch_bounds__(256) void sage_gemm_kernel(
    const float* __restrict__ feat,   // [nRows][128]
    const float* __restrict__ msg,    // [nRows][128]
    const float* __restrict__ deg,    // [nRows]
    const float* __restrict__ Wm,     // [256][128] row-major
    const float* __restrict__ bias,   // [128]
    float* __restrict__ out,          // [nRows][128]
    int nRows) {
  __shared__ float Wt[D_OUT * TSTRIDE];   // transposed chunk: Wt[n*68 + k]

  const int lane = threadIdx.x & 31;
  const int wave = threadIdx.x >> 5;
  const int lo   = lane & 15;   // M row (A) / N col (B,D) within tile
  const int hi   = lane >> 4;   // selects K pair (A,B) / M half (D)

  const int rowTile = blockIdx.x * 128 + wave * 16;
  const int rowA    = rowTile + lo;                       // A-row this lane feeds
  const int rowAc   = rowA < nRows ? rowA : (nRows - 1);  // clamped (EXEC stays full)

  float dg   = deg[rowAc];
  float rdeg = 1.0f / (dg > 1.0f ? dg : 1.0f);

  const float* featRow = feat + (size_t)rowAc * D_FEAT;
  const float* msgRow  = msg  + (size_t)rowAc * D_FEAT;

  // staging mapping: lane index -> k-row so LDS stores are bank-consecutive
  const int sr  = threadIdx.x & 63;        // k-row within chunk (0..63)
  const int sc0 = threadIdx.x >> 6;        // starting float4 column group (0..3)

  // per-lane base into transposed LDS for B fragments
  const float* wt = &Wt[lo * TSTRIDE + hi * 2];

  v8f acc[8] = {};  // 8 N-tiles of 16x16 f32

  for (int kc = 0; kc < K_TOTAL; kc += KC) {
    __syncthreads();  // previous chunk fully consumed
    // stage W[kc..kc+63][0..127] transposed into LDS.
    // Global: lane reads 16B of row (kc+sr), 512B apart across lanes (L2-hot).
    // LDS: stores at (4*c4+q)*68 + sr -> banks = lane + const: conflict-free.
#pragma unroll
    for (int i = 0; i < 8; ++i) {
      int c4 = sc0 + i * 4;                 // 0..31
      float4 wv = *(const float4*)(Wm + (size_t)(kc + sr) * D_OUT + c4 * 4);
      int c = c4 * 4;
      Wt[(c + 0) * TSTRIDE + sr] = wv.x;
      Wt[(c + 1) * TSTRIDE + sr] = wv.y;
      Wt[(c + 2) * TSTRIDE + sr] = wv.z;
      Wt[(c + 3) * TSTRIDE + sr] = wv.w;
    }
    __syncthreads();

    const float* xbase = ((kc < D_FEAT) ? (featRow + kc)
                                        : (msgRow + (kc - D_FEAT))) + hi * 2;
    const float scale  = (kc < D_FEAT) ? 1.0f : rdeg;

#pragma unroll
    for (int k4 = 0; k4 < 16; k4 += 2) {
      const int k = k4 * 4;

      // A fragments for both k-steps of this group
      v2f a0 = (*(const v2f*)(xbase + k))     * scale;
      v2f a1 = (*(const v2f*)(xbase + k + 4)) * scale;

      // batch-load ALL 16 B fragments before any WMMA (8x ds_load_2addr_b64)
      v2f b0[8], b1[8];
#pragma unroll
      for (int t = 0; t < 8; ++t) {
        b0[t] = *(const v2f*)(wt + t * (16 * TSTRIDE) + k);
        b1[t] = *(const v2f*)(wt + t * (16 * TSTRIDE) + k + 4);
      }

      // 16-WMMA burst
#pragma unroll
      for (int t = 0; t < 8; ++t) {
        acc[t] = __builtin_amdgcn_wmma_f32_16x16x4_f32(
            false, a0, false, b0[t], (short)0, acc[t], false, false);
        acc[t] = __builtin_amdgcn_wmma_f32_16x16x4_f32(
            false, a1, false, b1[t], (short)0, acc[t], false, false);
      }
    }
  }

  // Epilogue: D layout -> lane (lo = N col, hi picks M half), 8 rows per lane.
#pragma unroll
  for (int t = 0; t < 8; ++t) {
    int   n  = t * 16 + lo;
    float bv = bias[n];
#pragma unroll
    for (int r = 0; r < 8; ++r) {
      int orow = rowTile + hi * 8 + r;
      if (orow < nRows) out[(size_t)orow * D_OUT + n] = acc[t][r] + bv;
    }
  }
}

// ---------------------------------------------------------------------------
// Launcher
// ---------------------------------------------------------------------------
extern "C" void kernel_launch(void* const* d_in, const int* in_sizes, int n_in,
                              void* d_out, int out_size, void* d_ws, size_t ws_size,
                              hipStream_t stream) {
  (void)n_in; (void)out_size; (void)ws_size;

  const float* feat = (const float*)d_in[0];  // [N, 128]
  const int*   src  = (const int*)d_in[1];    // [E]
  const int*   dst  = (const int*)d_in[2];    // [E]
  const float* Wm   = (const float*)d_in[3];  // [256, 128]
  const float* bias = (const float*)d_in[4];  // [128]
  float*       out  = (float*)d_out;          // [N, 128]

  const int nNodes = in_sizes[0] / D_FEAT;
  const int nEdges = in_sizes[1];

  float* msg = (float*)d_ws;                      // [N, 128]
  float* deg = msg + (size_t)nNodes * D_FEAT;     // [N]

  int zn = nNodes * D_FEAT + nNodes;
  sage_zero_kernel<<<(zn + 255) / 256, 256, 0, stream>>>(msg, zn);

  long long threadsScatter = (long long)nEdges * 32;
  sage_scatter_kernel<<<(int)((threadsScatter + 255) / 256), 256, 0, stream>>>(
      feat, src, dst, msg, deg, nEdges);

  sage_gemm_kernel<<<(nNodes + 127) / 128, 256, 0, stream>>>(
      feat, msg, deg, Wm, bias, out, nNodes);
}